// MacroDGRCL_55825984913536
// MI455X (gfx1250) — compile-verified
//
#include <hip/hip_runtime.h>
#include <math.h>

#define N_S 50000
#define N_M 64
#define D_S 128
#define D_M 32
#define HH 4
#define DD 64
#define HD 256   /* H*D */
#define E_SS 800000
#define E_MS 200000
#define SLOPE 0.2f

typedef float v2f __attribute__((ext_vector_type(2)));
typedef float v8f __attribute__((ext_vector_type(8)));
typedef unsigned int u32v4 __attribute__((ext_vector_type(4)));
typedef int i32v8 __attribute__((ext_vector_type(8)));
typedef int i32v4 __attribute__((ext_vector_type(4)));

__device__ __forceinline__ void atomicMaxF(float* addr, float val) {
  // IEEE-754 ordered-int trick: works with -inf init, mixed signs.
  if (val >= 0.0f) atomicMax((int*)addr, __float_as_int(val));
  else             atomicMin((unsigned int*)addr, __float_as_uint(val));
}

// ---------------------------------------------------------------------------
// Generic fp32 WMMA GEMM: Dst(MxN) = A(MxK) * B(KxN) [+bias] [elu]
// One 16-row M-tile per block. The 16xK fp32 A-tile is DMA'd into LDS once
// per block by the Tensor Data Mover (TENSOR_LOAD_TO_LDS, TENSORcnt), then
// every wave feeds its V_WMMA_F32_16X16X4_F32 chain from ds_load.
// M, N multiples of 16; K multiple of 4 and <= 256.
// ---------------------------------------------------------------------------
__global__ void wmma_gemm_f32(const float* __restrict__ A, const float* __restrict__ B,
                              float* __restrict__ Dst, int M, int N, int K,
                              const float* __restrict__ bias, int doElu) {
  __shared__ float smemA[16 * 256];      // 16 KB, covers K up to 256

  const int mt    = blockIdx.x;
  const int nwaves= blockDim.x >> 5;
  const int wave  = threadIdx.x >> 5;
  const int lane  = threadIdx.x & 31;
  const int lh    = lane >> 4;          // which K-half (0: K=0..1, 1: K=2..3)
  const int l16   = lane & 15;
  const int ntile = N >> 4;

  // --- TDM: DMA the 16 x K fp32 A-tile (row stride K) into LDS. Wave 0 only.
  if (wave == 0) {
    unsigned long long gaddr =
        (unsigned long long)(uintptr_t)(A + (size_t)mt * 16 * K);
    unsigned ldsoff = (unsigned)(uintptr_t)(&smemA[0]);  // flat low32 = LDS byte offset

    u32v4 g0;
    g0[0] = 1u;                                            // count=1 (valid), user mode
    g0[1] = ldsoff;                                        // lds_addr
    g0[2] = (unsigned)(gaddr & 0xFFFFFFFFull);             // global_addr[31:0]
    g0[3] = (unsigned)((gaddr >> 32) & 0x01FFFFFFull)      // global_addr[56:32]
          | (2u << 30);                                    // type=2 ("image")

    i32v8 g1;
    g1[0] = (int)(2u << 16);                               // data_size=4B; mask=0; no pad
    g1[1] = (int)(((unsigned)K & 0xFFFFu) << 16);          // tensor_dim0[15:0] = K
    g1[2] = (int)((((unsigned)K >> 16) & 0xFFFFu)          // tensor_dim0[31:16]
          | (16u << 16));                                  // tensor_dim1[15:0] = 16
    g1[3] = (int)(((unsigned)K & 0xFFFFu) << 16);          // tensor_dim1 hi=0; tile_dim0=K
    g1[4] = 16;                                            // tile_dim1=16; tile_dim2=0
    g1[5] = K;                                             // tensor_dim0_stride[31:0]=K elems
    g1[6] = 0;                                             // stride hi; dim1_stride lo
    g1[7] = 0;                                             // dim1_stride hi

    i32v4 gz4 = {0, 0, 0, 0};                              // groups 2/3: 2D tile, unused
    i32v8 gz8 = {0, 0, 0, 0, 0, 0, 0, 0};                  // extra group (clang-23 form)
    // 6-arg amdgpu-toolchain form: (g0, g1, g2, g3, g?, cpol)
    __builtin_amdgcn_tensor_load_to_lds(g0, g1, gz4, gz4, gz8, 0);
    __builtin_amdgcn_s_wait_tensorcnt(0);
  }
  __syncthreads();

  for (int nt = wave; nt < ntile; nt += nwaves) {
    const int col = nt * 16 + l16;      // B/D column for this lane
    v8f c = {0.f,0.f,0.f,0.f,0.f,0.f,0.f,0.f};
    for (int k = 0; k < K; k += 4) {
      const int ka = k + lh * 2;
      v2f a, b;
      a.x = smemA[l16 * K + ka];        // A fragment from LDS (ds_load)
      a.y = smemA[l16 * K + ka + 1];
      b.x = B[(size_t)ka * N + col];
      b.y = B[(size_t)(ka + 1) * N + col];
      // (neg_a, A, neg_b, B, c_mod, C, reuse_a, reuse_b)
      c = __builtin_amdgcn_wmma_f32_16x16x4_f32(false, a, false, b,
                                                (short)0, c, false, false);
    }
#pragma unroll
    for (int r = 0; r < 8; ++r) {
      const int orow = mt * 16 + r + lh * 8;   // C/D layout: VGPR r -> M=r / M=8+r
      float v = c[r];
      if (bias)  v += bias[col];
      if (doElu) v = v > 0.f ? v : expm1f(v);
      Dst[(size_t)orow * N + col] = v;
    }
  }
}

// ---------------------------------------------------------------------------
// Per-node attention logits.
// ---------------------------------------------------------------------------
__global__ void stock_scores_kernel(const float* __restrict__ hs,
                                    const float* __restrict__ att_s,
                                    const float* __restrict__ att_m,
                                    float* __restrict__ si, float* __restrict__ sj,
                                    float* __restrict__ mi) {
  int idx = blockIdx.x * blockDim.x + threadIdx.x;
  if (idx >= N_S * HH) return;
  int n = idx >> 2, h = idx & 3;
  const float* row = hs + (size_t)n * HD + h * DD;
  const float* ai  = att_s + h * 2 * DD;        // att_stock[0,h,:D]
  const float* aj  = ai + DD;                   // att_stock[0,h,D:]
  const float* am  = att_m + h * 2 * DD;        // att_macro[0,h,:D]
  float s1 = 0.f, s2 = 0.f, s3 = 0.f;
#pragma unroll 4
  for (int d = 0; d < DD; ++d) {
    float v = row[d];
    s1 = fmaf(v, ai[d], s1);
    s2 = fmaf(v, aj[d], s2);
    s3 = fmaf(v, am[d], s3);
  }
  si[idx] = s1; sj[idx] = s2; mi[idx] = s3;
}

__global__ void macro_scores_kernel(const float* __restrict__ hm,
                                    const float* __restrict__ att_m,
                                    float* __restrict__ mj) {
  int idx = threadIdx.x;                 // one block of 256 covers 64*4
  if (idx >= N_M * HH) return;
  int n = idx >> 2, h = idx & 3;
  const float* row = hm + (size_t)n * HD + h * DD;
  const float* aj  = att_m + h * 2 * DD + DD;   // att_macro[0,h,D:]
  float s = 0.f;
  for (int d = 0; d < DD; ++d) s = fmaf(row[d], aj[d], s);
  mj[idx] = s;
}

// ---------------------------------------------------------------------------
// Segment softmax over edges: max pass, then exp/sum pass.
// ---------------------------------------------------------------------------
__global__ void edge_max_kernel(const float* __restrict__ sdst, const float* __restrict__ ssrc,
                                const int* __restrict__ src, const int* __restrict__ dst,
                                float* __restrict__ smax, int E) {
  int idx = blockIdx.x * blockDim.x + threadIdx.x;
  if (idx >= E * HH) return;
  int e = idx >> 2, h = idx & 3;
  float x = sdst[dst[e] * HH + h] + ssrc[src[e] * HH + h];
  x = x > 0.f ? x : SLOPE * x;           // leaky_relu
  atomicMaxF(&smax[dst[e] * HH + h], x);
}

__global__ void edge_exp_kernel(const float* __restrict__ sdst, const float* __restrict__ ssrc,
                                const int* __restrict__ src, const int* __restrict__ dst,
                                const float* __restrict__ smax, float* __restrict__ ssum,
                                float* __restrict__ ew, int E) {
  int idx = blockIdx.x * blockDim.x + threadIdx.x;
  if (idx >= E * HH) return;
  int e = idx >> 2, h = idx & 3;
  int d = dst[e];
  float x = sdst[d * HH + h] + ssrc[src[e] * HH + h];
  x = x > 0.f ? x : SLOPE * x;
  float w = expf(x - smax[d * HH + h]);
  ew[idx] = w;
  atomicAdd(&ssum[d * HH + h], w);
}

// ---------------------------------------------------------------------------
// Wave-per-edge gather/scale/scatter: agg[dst] += hsrc[src] * coeff(head).
// eweight == nullptr for macro edges (no edge weight there).
// ---------------------------------------------------------------------------
__global__ void edge_agg_kernel(const float* __restrict__ hsrc,
                                const float* __restrict__ ew, const float* __restrict__ ssum,
                                const float* __restrict__ eweight,
                                const int* __restrict__ src, const int* __restrict__ dst,
                                float* __restrict__ agg, int E) {
  int gtid = blockIdx.x * blockDim.x + threadIdx.x;
  int e    = gtid >> 5;
  int lane = threadIdx.x & 31;
  if (e >= E) return;
  int s = src[e], d = dst[e];
  float wedge = eweight ? eweight[e] : 1.0f;
  const float* hrow = hsrc + (size_t)s * HD;
  float*       arow = agg  + (size_t)d * HD;
  float coeff[HH];
#pragma unroll
  for (int h = 0; h < HH; ++h)
    coeff[h] = ew[(size_t)e * HH + h] / (ssum[d * HH + h] + 1e-16f) * wedge;
#pragma unroll
  for (int i = 0; i < 8; ++i) {          // 8 * 32 lanes = 256 values, coalesced
    int v = lane + i * 32;
    atomicAdd(&arow[v], hrow[v] * coeff[v >> 6]);
  }
}

// ---------------------------------------------------------------------------
// Init kernels (agg must be zeroed, softmax stats re-armed every call).
// ---------------------------------------------------------------------------
__global__ void zero_kernel(float* __restrict__ p, size_t n) {
  size_t i = (size_t)blockIdx.x * blockDim.x + threadIdx.x;
  if (i < n) p[i] = 0.f;
}

__global__ void init_stats_kernel(float* __restrict__ smax_s, float* __restrict__ ssum_s,
                                  float* __restrict__ smax_m, float* __restrict__ ssum_m) {
  int i = blockIdx.x * blockDim.x + threadIdx.x;
  if (i >= N_S * HH) return;
  smax_s[i] = -INFINITY; ssum_s[i] = 0.f;
  smax_m[i] = -INFINITY; ssum_m[i] = 0.f;
}

extern "C" void kernel_launch(void* const* d_in, const int* in_sizes, int n_in,
                              void* d_out, int out_size, void* d_ws, size_t ws_size,
                              hipStream_t stream) {
  const float* stock_h     = (const float*)d_in[0];
  const float* macro_h     = (const float*)d_in[1];
  const float* edge_weight = (const float*)d_in[2];
  const int*   ss_src      = (const int*)d_in[3];
  const int*   ss_dst      = (const int*)d_in[4];
  const int*   ms_src      = (const int*)d_in[5];
  const int*   ms_dst      = (const int*)d_in[6];
  const float* W_stock     = (const float*)d_in[7];
  const float* W_macro     = (const float*)d_in[8];
  const float* att_stock   = (const float*)d_in[9];
  const float* att_macro   = (const float*)d_in[10];
  const float* Wo          = (const float*)d_in[11];
  const float* bo          = (const float*)d_in[12];
  float* out = (float*)d_out;

  // Workspace carve-up (~124 MB of floats).
  float* ws = (float*)d_ws;
  size_t off = 0;
  float* hs     = ws + off; off += (size_t)N_S * HD;   // 12.8M
  float* hm     = ws + off; off += (size_t)N_M * HD;
  float* agg    = ws + off; off += (size_t)N_S * HD;   // 12.8M
  float* si     = ws + off; off += (size_t)N_S * HH;
  float* sj     = ws + off; off += (size_t)N_S * HH;
  float* mi     = ws + off; off += (size_t)N_S * HH;
  float* mj     = ws + off; off += (size_t)N_M * HH;
  float* smax_s = ws + off; off += (size_t)N_S * HH;
  float* ssum_s = ws + off; off += (size_t)N_S * HH;
  float* smax_m = ws + off; off += (size_t)N_S * HH;
  float* ssum_m = ws + off; off += (size_t)N_S * HH;
  float* ew_ss  = ws + off; off += (size_t)E_SS * HH;  // 3.2M
  float* ew_ms  = ws + off; off += (size_t)E_MS * HH;

  // 0) init accumulators
  {
    size_t n = (size_t)N_S * HD;
    zero_kernel<<<(unsigned)((n + 255) / 256), 256, 0, stream>>>(agg, n);
    init_stats_kernel<<<(N_S * HH + 255) / 256, 256, 0, stream>>>(smax_s, ssum_s, smax_m, ssum_m);
  }
  // 1) projections (fp32 WMMA, TDM-staged A tiles)
  wmma_gemm_f32<<<N_S / 16, 256, 0, stream>>>(stock_h, W_stock, hs, N_S, HD, D_S, nullptr, 0);
  wmma_gemm_f32<<<N_M / 16, 256, 0, stream>>>(macro_h, W_macro, hm, N_M, HD, D_M, nullptr, 0);
  // 2) attention logits
  stock_scores_kernel<<<(N_S * HH + 255) / 256, 256, 0, stream>>>(hs, att_stock, att_macro, si, sj, mi);
  macro_scores_kernel<<<1, 256, 0, stream>>>(hm, att_macro, mj);
  // 3) stock-stock edges: softmax + weighted aggregate
  edge_max_kernel<<<(E_SS * HH + 255) / 256, 256, 0, stream>>>(si, sj, ss_src, ss_dst, smax_s, E_SS);
  edge_exp_kernel<<<(E_SS * HH + 255) / 256, 256, 0, stream>>>(si, sj, ss_src, ss_dst, smax_s, ssum_s, ew_ss, E_SS);
  edge_agg_kernel<<<(E_SS * 32 + 255) / 256, 256, 0, stream>>>(hs, ew_ss, ssum_s, edge_weight, ss_src, ss_dst, agg, E_SS);
  // 4) macro-stock edges
  edge_max_kernel<<<(E_MS * HH + 255) / 256, 256, 0, stream>>>(mi, mj, ms_src, ms_dst, smax_m, E_MS);
  edge_exp_kernel<<<(E_MS * HH + 255) / 256, 256, 0, stream>>>(mi, mj, ms_src, ms_dst, smax_m, ssum_m, ew_ms, E_MS);
  edge_agg_kernel<<<(E_MS * 32 + 255) / 256, 256, 0, stream>>>(hm, ew_ms, ssum_m, nullptr, ms_src, ms_dst, agg, E_MS);
  // 5) output projection + bias + ELU (fp32 WMMA)
  wmma_gemm_f32<<<N_S / 16, 128, 0, stream>>>(agg, Wo, out, N_S, DD, HD, bo, 1);
}